// Predicting_Base_21990232555637
// MI455X (gfx1250) — compile-verified
//
#include <hip/hip_runtime.h>
#include <hip/hip_bf16.h>
#include <stdint.h>

#define B_ 8
#define N_ 40000
#define C_ 80
#define K_ 500
#define M_ 4000          // B_*K_
#define THR_CONF_ 0.05f
#define THR_NMS_ 0.5f

// ---------------------------------------------------------------------------
// Stage 1: per-anchor max/argmax over 80 logits, sigmoid once per anchor.
// HBM-bound (102.4 MB @ 23.3 TB/s ~ 4.4us). Tiles are staged into LDS with the
// CDNA5 async VMEM->LDS path (global_load_async_to_lds_b128, ASYNCcnt), then
// reduced from a padded-pitch LDS layout with ds_load_b128 reads
// (pitch 84 floats: rows 16B-aligned, 4-way worst-case bank conflict).
// ---------------------------------------------------------------------------
#define APB 128     // anchors per block (== threads per block)
#define PITCH 84    // padded floats per anchor row in LDS
#define CH4 (C_ / 4)   // 20 float4 chunks per anchor

typedef int v4i __attribute__((ext_vector_type(4)));
typedef __attribute__((address_space(1))) v4i* gptr_v4i;
typedef __attribute__((address_space(3))) v4i* lptr_v4i;

__global__ __launch_bounds__(APB) void score_label_kernel(
    const float* __restrict__ pconf,
    float* __restrict__ pscores, int* __restrict__ plabels)
{
    __shared__ float tile[APB * PITCH];   // 43,008 B
    const long long blockAnchor = (long long)blockIdx.x * APB;
    const float* gbase = pconf + blockAnchor * C_;

#if __has_builtin(__builtin_amdgcn_global_load_async_to_lds_b128)
    // 2560 16B chunks per tile; thread t takes chunks t, t+128, ...
    // Chunk k = anchor*20 + c4: maintain (anchor, c4) incrementally.
    {
        int a  = threadIdx.x / CH4;       // 128/20: a in [0,6], c4 in [0,19]
        int c4 = threadIdx.x % CH4;
        #pragma unroll
        for (int it = 0; it < APB * CH4 / APB; ++it) {   // 20 iterations
            const float* g = gbase + (a * C_ + c4 * 4);
            float* l = &tile[a * PITCH + c4 * 4];
            __builtin_amdgcn_global_load_async_to_lds_b128(
                (gptr_v4i)g, (lptr_v4i)l, 0, 0);
            // advance by 128 chunks: 128 = 6*20 + 8
            a += 6; c4 += 8;
            if (c4 >= CH4) { c4 -= CH4; a += 1; }
        }
    }
#if __has_builtin(__builtin_amdgcn_s_wait_asynccnt)
    __builtin_amdgcn_s_wait_asynccnt(0);
#endif
#else
    // Fallback: plain float4 staging (keeps the file compiling if the async
    // builtins are absent on this toolchain).
    const float4* g4 = (const float4*)gbase;
    for (int k = threadIdx.x; k < APB * CH4; k += APB) {
        int a  = k / CH4;
        int c4 = k % CH4;
        *(float4*)&tile[a * PITCH + c4 * 4] = g4[k];
    }
#endif
    __syncthreads();

    const int a = threadIdx.x;
    const float4* row4 = (const float4*)&tile[a * PITCH];   // 336B offset, 16B aligned
    float best = -3.402823466e+38f;
    int   bi   = 0;
    #pragma unroll
    for (int c4 = 0; c4 < CH4; ++c4) {
        float4 v = row4[c4];              // ds_load_b128
        // strict '>' keeps the first (lowest) index on ties, matching jnp.argmax
        if (v.x > best) { best = v.x; bi = c4 * 4 + 0; }
        if (v.y > best) { best = v.y; bi = c4 * 4 + 1; }
        if (v.z > best) { best = v.z; bi = c4 * 4 + 2; }
        if (v.w > best) { best = v.w; bi = c4 * 4 + 3; }
    }
    // sigmoid(max(logits)) == max(sigmoid(logits)) (sigmoid is monotonic)
    float s = 1.0f / (1.0f + expf(-best));
    long long idx = blockAnchor + a;
    pscores[idx] = s;
    plabels[idx] = bi;
}

// ---------------------------------------------------------------------------
// Stage 2: exact top-500 per batch via two-level radix histogram on float bits
// (all scores >= 0 so IEEE bits are order-preserving), then an in-LDS bitonic
// sort of <=1024 candidates. Key = (scorebits << 32) | ~index : descending key
// order == (score desc, index asc), matching jax.lax.top_k.
// ---------------------------------------------------------------------------
__global__ __launch_bounds__(256) void topk_kernel(
    const float* __restrict__ pscores,
    float* __restrict__ scores_k, int* __restrict__ ids_k)
{
    const int b = blockIdx.x;
    const int tid = threadIdx.x;
    const float* sc = pscores + (long long)b * N_;

    __shared__ unsigned hist[8192];
    __shared__ unsigned part[256];
    __shared__ unsigned long long cand[1024];
    __shared__ unsigned sh_cut1, sh_above1, sh_cut2, sh_cnt;

    // ---- pass A: coarse histogram on bits[30:18] ----
    for (int i = tid; i < 8192; i += 256) hist[i] = 0;
    __syncthreads();
    for (int i = tid; i < N_; i += 256) {
        unsigned bits = __float_as_uint(sc[i]);
        atomicAdd(&hist[bits >> 18], 1u);
    }
    __syncthreads();
    { unsigned s = 0; for (int i = 0; i < 32; ++i) s += hist[tid * 32 + i]; part[tid] = s; }
    __syncthreads();
    if (tid == 0) {
        unsigned cum = 0; int chunk = 0;
        for (int t = 255; t >= 0; --t) {
            if (cum + part[t] >= K_) { chunk = t; break; }
            cum += part[t];
        }
        unsigned cut = 0, above = cum;
        for (int bin = chunk * 32 + 31; bin >= chunk * 32; --bin) {
            if (above + hist[bin] >= K_) { cut = (unsigned)bin; break; }
            above += hist[bin];
        }
        sh_cut1 = cut; sh_above1 = above;
    }
    __syncthreads();
    const unsigned cut1 = sh_cut1, above1 = sh_above1;

    // ---- pass B: refine inside the cut bin on bits[17:5] ----
    for (int i = tid; i < 8192; i += 256) hist[i] = 0;
    __syncthreads();
    for (int i = tid; i < N_; i += 256) {
        unsigned bits = __float_as_uint(sc[i]);
        if ((bits >> 18) == cut1) atomicAdd(&hist[(bits >> 5) & 8191], 1u);
    }
    __syncthreads();
    { unsigned s = 0; for (int i = 0; i < 32; ++i) s += hist[tid * 32 + i]; part[tid] = s; }
    __syncthreads();
    if (tid == 0) {
        unsigned cum = above1, cut = 0;
        for (int t = 255; t >= 0; --t) {
            if (cum + part[t] >= K_) {
                for (int bin = t * 32 + 31; bin >= t * 32; --bin) {
                    if (cum + hist[bin] >= K_) { cut = (unsigned)bin; break; }
                    cum += hist[bin];
                }
                break;
            }
            cum += part[t];
        }
        sh_cut2 = cut; sh_cnt = 0;
    }
    __syncthreads();
    const unsigned cut2 = sh_cut2;

    // ---- pass C: collect candidates (>=500, ~<=520 for continuous scores) ----
    for (int i = tid; i < N_; i += 256) {
        unsigned bits = __float_as_uint(sc[i]);
        unsigned b1 = bits >> 18;
        bool sel = (b1 > cut1) || (b1 == cut1 && ((bits >> 5) & 8191) >= cut2);
        if (sel) {
            unsigned p = atomicAdd(&sh_cnt, 1u);
            if (p < 1024)
                cand[p] = ((unsigned long long)bits << 32) | (unsigned)(~(unsigned)i);
        }
    }
    __syncthreads();
    for (unsigned p = sh_cnt + tid; p < 1024; p += 256) cand[p] = 0ull;
    __syncthreads();

    // ---- bitonic sort (ascending) of 1024 u64 keys in LDS ----
    for (unsigned sz = 2; sz <= 1024; sz <<= 1) {
        for (unsigned stride = sz >> 1; stride > 0; stride >>= 1) {
            for (unsigned t = tid; t < 512; t += 256) {
                unsigned pos = 2 * t - (t & (stride - 1));
                unsigned long long a0 = cand[pos], a1 = cand[pos + stride];
                bool up = ((pos & sz) == 0);
                if ((a0 > a1) == up) { cand[pos] = a1; cand[pos + stride] = a0; }
            }
            __syncthreads();
        }
    }

    // ---- emit top-K in descending order ----
    for (int r = tid; r < K_; r += 256) {
        unsigned long long key = cand[1023 - r];
        scores_k[b * K_ + r] = __uint_as_float((unsigned)(key >> 32));
        ids_k[b * K_ + r]    = (int)(~(unsigned)key);
    }
}

// ---------------------------------------------------------------------------
// Stage 3: gather boxes/labels for the top-K ids and write all fixed outputs.
// keep[] is zero-initialized here; NMS sets the kept entries to 1.
// ---------------------------------------------------------------------------
__global__ __launch_bounds__(256) void gather_kernel(
    const float* __restrict__ pboxes, const int* __restrict__ plabels,
    const float* __restrict__ scores_k, const int* __restrict__ ids_k,
    int* __restrict__ labels_g, float* __restrict__ out)
{
    int g = blockIdx.x * 256 + threadIdx.x;
    if (g >= M_) return;
    int b   = g / K_;
    int idx = ids_k[g];
    float s = scores_k[g];
    int lbl = plabels[(long long)b * N_ + idx];
    const float* bx = pboxes + ((long long)b * N_ + idx) * 4;
    out[g]              = (float)b;     // ids_batch1
    out[M_ + 4 * g + 0] = bx[0];        // p_boxes1
    out[M_ + 4 * g + 1] = bx[1];
    out[M_ + 4 * g + 2] = bx[2];
    out[M_ + 4 * g + 3] = bx[3];
    out[5 * M_ + g]     = (float)lbl;   // p_labels1
    out[6 * M_ + g]     = s;            // p_scores1
    out[7 * M_ + g]     = 0.0f;         // keep (default false)
    labels_g[g]         = lbl;
}

// ---------------------------------------------------------------------------
// Stage 4: greedy NMS, decomposed into 640 independent (batch,label) groups
// (the class-offset trick in the reference makes cross-group IoU exactly 0).
// One wave32 per group; candidates arrive in descending-score order already.
// ---------------------------------------------------------------------------
__global__ __launch_bounds__(32) void nms_kernel(
    const int* __restrict__ labels_g, float* __restrict__ out)
{
    const int batch = blockIdx.x & (B_ - 1);
    const int label = blockIdx.x >> 3;
    const int lane  = threadIdx.x;
    __shared__ float kx1[K_], ky1[K_], kx2[K_], ky2[K_];
    int cnt = 0;

    for (int r = 0; r < K_; ++r) {
        int g = batch * K_ + r;
        if (labels_g[g] != label) continue;          // uniform across wave
        float s = out[6 * M_ + g];
        if (!(s > THR_CONF_)) continue;              // invalid: keep stays 0
        float bx1 = out[M_ + 4 * g + 0];
        float by1 = out[M_ + 4 * g + 1];
        float bx2 = out[M_ + 4 * g + 2];
        float by2 = out[M_ + 4 * g + 3];
        float areaA = (bx2 - bx1) * (by2 - by1);

        bool sup = false;
        for (int base = 0; base < cnt && !sup; base += 32) {
            int k = base + lane;
            bool over = false;
            if (k < cnt) {
                float xx1 = fmaxf(bx1, kx1[k]);
                float yy1 = fmaxf(by1, ky1[k]);
                float xx2 = fminf(bx2, kx2[k]);
                float yy2 = fminf(by2, ky2[k]);
                float w = fmaxf(xx2 - xx1, 0.0f);
                float h = fmaxf(yy2 - yy1, 0.0f);
                float inter = w * h;
                float areaB = (kx2[k] - kx1[k]) * (ky2[k] - ky1[k]);
                float uni = areaA + areaB - inter;
                over = (inter / fmaxf(uni, 1e-9f)) > THR_NMS_;
            }
            if (__any(over)) sup = true;
        }
        if (!sup) {
            if (lane == 0) {
                kx1[cnt] = bx1; ky1[cnt] = by1; kx2[cnt] = bx2; ky2[cnt] = by2;
                out[7 * M_ + g] = 1.0f;
            }
            ++cnt;
        }
        __syncthreads();
    }
}

// ---------------------------------------------------------------------------
extern "C" void kernel_launch(void* const* d_in, const int* in_sizes, int n_in,
                              void* d_out, int out_size, void* d_ws, size_t ws_size,
                              hipStream_t stream)
{
    const float* pconf  = (const float*)d_in[0];   // (8,40000,80) f32
    const float* pboxes = (const float*)d_in[1];   // (8,40000,4)  f32
    float* out = (float*)d_out;                    // 32000 f32
    char* ws = (char*)d_ws;

    float* pscores  = (float*)(ws);                 // 320000 f32  (1,280,000 B)
    int*   plabels  = (int*)  (ws + 1280000);       // 320000 i32
    float* scores_k = (float*)(ws + 2560000);       // 4000 f32
    int*   ids_k    = (int*)  (ws + 2576000);       // 4000 i32
    int*   labels_g = (int*)  (ws + 2592000);       // 4000 i32

    score_label_kernel<<<(B_ * N_) / APB, APB, 0, stream>>>(pconf, pscores, plabels);
    topk_kernel<<<B_, 256, 0, stream>>>(pscores, scores_k, ids_k);
    gather_kernel<<<(M_ + 255) / 256, 256, 0, stream>>>(pboxes, plabels, scores_k,
                                                        ids_k, labels_g, out);
    nms_kernel<<<B_ * C_, 32, 0, stream>>>(labels_g, out);
}